// KGCN_32564442038934
// MI455X (gfx1250) — compile-verified
//
#include <hip/hip_runtime.h>
#include <hip/hip_bf16.h>

#define DIM    512
#define N_NBR  8
#define N_CTX  16
#define BATCH  1024

typedef __attribute__((ext_vector_type(2))) float v2f;
typedef __attribute__((ext_vector_type(8))) float v8f;
typedef int v4i __attribute__((vector_size(16)));

// ---- CDNA5 async global->LDS copy (ASYNCcnt path), with portable fallback ----
#if __has_builtin(__builtin_amdgcn_global_load_async_to_lds_b128)
__device__ __forceinline__ void async_b128(const void* g, void* l) {
    __builtin_amdgcn_global_load_async_to_lds_b128(
        (__attribute__((address_space(1))) v4i*)(g),
        (__attribute__((address_space(3))) v4i*)(l), 0, 0);
}
#define ASYNC_WAIT() asm volatile("s_wait_asynccnt 0" ::: "memory")
#else
__device__ __forceinline__ void async_b128(const void* g, void* l) {
    *(float4*)(l) = *(const float4*)(g);
}
#define ASYNC_WAIT() ((void)0)
#endif

// ---------------------------------------------------------------------------
// Kernel A: embedding gathers + attention-weighted neighbor aggregation.
// One block per batch element, 256 threads; each thread owns dims d and d+256.
// Writes hidden[b,d] = self_vec + agg  and caches user[b,d] to linear scratch.
// ---------------------------------------------------------------------------
__global__ void kgcn_gather_attn(const int* __restrict__ u,
                                 const int* __restrict__ v,
                                 const int* __restrict__ adj_ent,
                                 const int* __restrict__ adj_rel,
                                 const float* __restrict__ usr_emb,
                                 const float* __restrict__ ent_emb,
                                 const float* __restrict__ rel_emb,
                                 float* __restrict__ hidden,
                                 float* __restrict__ user_s) {
    const int b   = blockIdx.x;
    const int tid = threadIdx.x;          // 0..255

    __shared__ float red[N_NBR * 256];    // 8 KB partial dot products
    __shared__ float wsm[N_NBR];          // softmax weights
    __shared__ int   nbr[N_NBR];
    __shared__ int   rel[N_NBR];

    const int uid = u[b];
    const int vid = v[b];

    if (tid < N_NBR)               nbr[tid]     = adj_ent[vid * N_NBR + tid];
    if (tid >= N_NBR && tid < 16)  rel[tid - 8] = adj_rel[vid * N_NBR + (tid - 8)];
    __syncthreads();

    const float u0 = usr_emb[(size_t)uid * DIM + tid];
    const float u1 = usr_emb[(size_t)uid * DIM + 256 + tid];
    user_s[(size_t)b * DIM + tid]       = u0;
    user_s[(size_t)b * DIM + 256 + tid] = u1;

    // per-thread partial attention scores: dot(user, rel_emb[nbr_r[k]])
#pragma unroll
    for (int k = 0; k < N_NBR; ++k) {
        const float* rv = rel_emb + (size_t)rel[k] * DIM;
        red[k * 256 + tid] = u0 * rv[tid] + u1 * rv[256 + tid];
    }
    __syncthreads();

    // block tree-reduction for all 8 scores simultaneously
    for (int stride = 128; stride > 0; stride >>= 1) {
        if (tid < stride) {
#pragma unroll
            for (int k = 0; k < N_NBR; ++k)
                red[k * 256 + tid] += red[k * 256 + tid + stride];
        }
        __syncthreads();
    }

    if (tid == 0) {
        float mx = red[0];
#pragma unroll
        for (int k = 1; k < N_NBR; ++k) mx = fmaxf(mx, red[k * 256]);
        float s = 0.f;
        float e[N_NBR];
#pragma unroll
        for (int k = 0; k < N_NBR; ++k) { e[k] = __expf(red[k * 256] - mx); s += e[k]; }
        const float inv = 1.f / s;
#pragma unroll
        for (int k = 0; k < N_NBR; ++k) wsm[k] = e[k] * inv;
    }
    __syncthreads();

    float agg0 = 0.f, agg1 = 0.f;
#pragma unroll
    for (int k = 0; k < N_NBR; ++k) {
        const float  wk = wsm[k];
        const float* nv = ent_emb + (size_t)nbr[k] * DIM;
        agg0 += wk * nv[tid];
        agg1 += wk * nv[256 + tid];
    }
    const float* sv = ent_emb + (size_t)vid * DIM;
    hidden[(size_t)b * DIM + tid]       = sv[tid]       + agg0;
    hidden[(size_t)b * DIM + 256 + tid] = sv[256 + tid] + agg1;
}

// ---------------------------------------------------------------------------
// Kernel B: item = tanh(hidden @ W_agg^T + b_agg)   [1024x512]x[512x512]
// f32 WMMA (V_WMMA_F32_16X16X4_F32) fed from double-buffered LDS panels
// staged with GLOBAL_LOAD_ASYNC_TO_LDS_B128 (ASYNCcnt): stage s+1's DMA
// streams in while stage s's 16 WMMAs execute.
//
// Block = 256 thr / 8 waves, macro-tile BM=64 x BN=32 (4x2 WMMA tiles).
// LDS rows padded to 68 dwords: 16B-aligned B128 writes + conflict-free
// ds_load_b64 fragment reads (bank = (4*row + k) mod 64).
// ---------------------------------------------------------------------------
#define BM  64
#define BN  32
#define KC  64
#define LDK (KC + 4)   // padded row stride in dwords

__global__ void kgcn_gemm_agg(const float* __restrict__ hidden,
                              const float* __restrict__ W_agg,
                              const float* __restrict__ b_agg,
                              float* __restrict__ item) {
    __shared__ float As[2][BM * LDK];   // 2 x 17.4 KB
    __shared__ float Bs[2][BN * LDK];   // 2 x  8.7 KB

    const int tid  = threadIdx.x;
    const int wave = tid >> 5;
    const int lane = tid & 31;
    const int l15  = lane & 15;
    const int hi   = lane >> 4;                 // k-pair select

    const int bm0 = (blockIdx.x >> 4) * BM;     // 16 blocks along M (batch)
    const int bn0 = (blockIdx.x & 15) * BN;     // 16 blocks along N (out dim)
    const int wm  = wave >> 1;                  // 0..3
    const int wn  = wave & 1;                   // 0..1

    // per-thread copy slices (A: 4 x B128, B: 2 x B128)
    const int arow0 = tid >> 4;                 // s = tid + i*256 -> row = s/16
    const int seg   = tid & 15;

    auto stage_copy = [&](int kb, int buf) {
#pragma unroll
        for (int i = 0; i < 4; ++i) {
            const int row = arow0 + i * 16;
            async_b128(hidden + (size_t)(bm0 + row) * DIM + kb + seg * 4,
                       &As[buf][row * LDK + seg * 4]);
        }
#pragma unroll
        for (int i = 0; i < 2; ++i) {
            const int row = arow0 + i * 16;     // 0..31
            async_b128(W_agg + (size_t)(bn0 + row) * DIM + kb + seg * 4,
                       &Bs[buf][row * LDK + seg * 4]);
        }
    };

    v8f acc = {};

    stage_copy(0, 0);                          // prologue: stage 0 in flight

    const int NSTAGE = DIM / KC;               // 8
    for (int s = 0; s < NSTAGE; ++s) {
        const int cur = s & 1;
        ASYNC_WAIT();          // drain this wave's stage-s copies
        __syncthreads();       // panel s visible; buf[cur^1] free for overwrite

        if (s + 1 < NSTAGE)
            stage_copy((s + 1) * KC, cur ^ 1);   // overlap DMA with compute

        const float* Arow = &As[cur][(wm * 16 + l15) * LDK];
        const float* Brow = &Bs[cur][(wn * 16 + l15) * LDK];
#pragma unroll
        for (int k = 0; k < KC; k += 4) {
            v2f a, bf;
            a.x  = Arow[k + 2 * hi];
            a.y  = Arow[k + 2 * hi + 1];
            bf.x = Brow[k + 2 * hi];
            bf.y = Brow[k + 2 * hi + 1];
            acc = __builtin_amdgcn_wmma_f32_16x16x4_f32(
                false, a, false, bf, (short)0, acc, false, false);
        }
        // next iteration's ASYNC_WAIT + barrier protects buf[cur] before its
        // reuse at stage s+2, so no trailing barrier needed here.
    }

    const int   n    = bn0 + wn * 16 + l15;
    const float bias = b_agg[n];
#pragma unroll
    for (int r = 0; r < 8; ++r) {
        const int m = bm0 + wm * 16 + r + 8 * hi;
        item[(size_t)m * DIM + n] = tanhf(acc[r] + bias);
    }
}

// ---------------------------------------------------------------------------
// Kernel C: c[t, d] = sum_b item[b, d] * W_lin[t, b] + b_lin[t]   [16 x 512]
// M=16 (N_CTX), N=512, K=1024 (batch). 32 tiles, f32 WMMA, direct loads.
// ---------------------------------------------------------------------------
__global__ void kgcn_gemm_lin(const float* __restrict__ item,
                              const float* __restrict__ W_lin,
                              const float* __restrict__ b_lin,
                              float* __restrict__ c_out) {
    const int wave = threadIdx.x >> 5;
    const int lane = threadIdx.x & 31;
    const int tn   = blockIdx.x * (blockDim.x >> 5) + wave;   // 0..31
    const int l15  = lane & 15;
    const int hi   = lane >> 4;

    const float* Arow = W_lin + (size_t)l15 * BATCH;          // W_lin[t][*]
    const int    n    = tn * 16 + l15;

    v8f acc = {};
#pragma unroll 8
    for (int k = 0; k < BATCH; k += 4) {
        v2f a, bf;
        a.x  = Arow[k + 2 * hi];
        a.y  = Arow[k + 2 * hi + 1];
        bf.x = item[(size_t)(k + 2 * hi) * DIM + n];          // B[k][n] = item[b][d]
        bf.y = item[(size_t)(k + 2 * hi + 1) * DIM + n];
        acc = __builtin_amdgcn_wmma_f32_16x16x4_f32(
            false, a, false, bf, (short)0, acc, false, false);
    }

#pragma unroll
    for (int r = 0; r < 8; ++r) {
        const int t = r + 8 * hi;                             // 0..15
        c_out[(size_t)t * DIM + n] = acc[r] + b_lin[t];
    }
}

// ---------------------------------------------------------------------------
// Kernel D: out[b] = sigmoid(dot(user[b], item[b]))
// ---------------------------------------------------------------------------
__global__ void kgcn_dot_sig(const float* __restrict__ user_s,
                             const float* __restrict__ item,
                             float* __restrict__ out) {
    const int b   = blockIdx.x;
    const int tid = threadIdx.x;   // 0..255
    __shared__ float red[256];

    const float* uvec = user_s + (size_t)b * DIM;
    const float* ivec = item   + (size_t)b * DIM;
    red[tid] = uvec[tid] * ivec[tid] + uvec[256 + tid] * ivec[256 + tid];
    __syncthreads();
    for (int stride = 128; stride > 0; stride >>= 1) {
        if (tid < stride) red[tid] += red[tid + stride];
        __syncthreads();
    }
    if (tid == 0) out[b] = 1.f / (1.f + __expf(-red[0]));
}

// ---------------------------------------------------------------------------
extern "C" void kernel_launch(void* const* d_in, const int* in_sizes, int n_in,
                              void* d_out, int out_size, void* d_ws, size_t ws_size,
                              hipStream_t stream) {
    const int*   u       = (const int*)  d_in[0];
    const int*   v       = (const int*)  d_in[1];
    const int*   adj_ent = (const int*)  d_in[2];
    const int*   adj_rel = (const int*)  d_in[3];
    const float* usr_emb = (const float*)d_in[4];
    const float* ent_emb = (const float*)d_in[5];
    const float* rel_emb = (const float*)d_in[6];
    const float* W_agg   = (const float*)d_in[7];
    const float* b_agg   = (const float*)d_in[8];
    const float* W_lin   = (const float*)d_in[9];
    const float* b_lin   = (const float*)d_in[10];

    float* out_sig = (float*)d_out;            // [1024]
    float* c_out   = out_sig + BATCH;          // [16*512]

    float* hidden = (float*)d_ws;                          // [1024*512]
    float* user_s = hidden + (size_t)BATCH * DIM;          // [1024*512]
    float* item   = user_s + (size_t)BATCH * DIM;          // [1024*512]

    kgcn_gather_attn<<<BATCH, 256, 0, stream>>>(u, v, adj_ent, adj_rel,
                                                usr_emb, ent_emb, rel_emb,
                                                hidden, user_s);

    // 16x16 macro-tile grid (BM=64 x BN=32) -> 256 blocks, 8 waves each
    kgcn_gemm_agg<<<256, 256, 0, stream>>>(hidden, W_agg, b_agg, item);

    // 32 tiles, 8 waves per block -> 4 blocks
    kgcn_gemm_lin<<<4, 256, 0, stream>>>(item, W_lin, b_lin, c_out);

    kgcn_dot_sig<<<BATCH, 256, 0, stream>>>(user_s, item, out_sig);
}